// TexSphere_19146964205931
// MI455X (gfx1250) — compile-verified
//
#include <hip/hip_runtime.h>

typedef float v2f __attribute__((ext_vector_type(2)));
typedef float v8f __attribute__((ext_vector_type(8)));

#define NLEV 16
#define TMASK ((1u << 19) - 1u)
#define P1 2654435761u
#define P2 805459861u
#define PAD 34   // floats per point row in LDS: 32 feats + 2 pad (8B aligned, bank-conflict free)

// res[lv] = floor(16 * (2048/16)^(lv/15)) computed in fp64, matches f32 reference values
__constant__ float c_res[NLEV] = {
    16.f, 22.f, 30.f, 42.f, 58.f, 80.f, 111.f, 154.f,
    212.f, 294.f, 406.f, 561.f, 776.f, 1072.f, 1482.f, 2048.f};

__global__ __launch_bounds__(256) void hashgrid_wmma_kernel(
    const float* __restrict__ xin,
    const float* __restrict__ table,
    const float* __restrict__ W,     // [3, 32]
    const float* __restrict__ bias,  // [3]
    float* __restrict__ out,         // [N, 3]
    int N)
{
    __shared__ float sFeat[256 * PAD];   // per-point 32 features, padded stride
    __shared__ v2f   sB[8 * 32];         // B fragments for 8 K-chunks of V_WMMA_F32_16X16X4_F32
    __shared__ float sOut[256 * 3];      // D staging for coalesced output stores

    const int tid = threadIdx.x;
    const int gid = blockIdx.x * 256 + tid;
    const int pid = (gid < N) ? gid : (N - 1);   // clamp so EXEC stays full for WMMA

    // ---- Build B fragments (W^T padded to 4x16 per chunk), 256 threads -> 8*32 elems ----
    {
        const int j = tid >> 5;          // K-chunk 0..7
        const int l = tid & 31;          // lane within fragment
        const int n = l & 15;            // output column
        const int kb = 4 * j + ((l >= 16) ? 2 : 0);
        v2f bb;
        bb.x = (n < 3) ? W[n * 32 + kb]     : 0.f;   // K = kb   row
        bb.y = (n < 3) ? W[n * 32 + kb + 1] : 0.f;   // K = kb+1 row
        sB[tid] = bb;
    }

    // ---- Phase 1: hash-grid encode, one point per thread ----
    const float px = xin[pid * 3 + 0];
    const float py = xin[pid * 3 + 1];
    const float pz = xin[pid * 3 + 2];
    const float xn0 = (px + 3.f) * (1.f / 6.f);
    const float xn1 = (py + 3.f) * (1.f / 6.f);
    const float xn2 = (pz + 3.f) * (1.f / 6.f);

    float* myFeat = &sFeat[tid * PAD];

    #pragma unroll
    for (int lv = 0; lv < NLEV; ++lv) {
        const float r = c_res[lv];
        const float fx = xn0 * r, fy = xn1 * r, fz = xn2 * r;
        const float f0x = floorf(fx), f0y = floorf(fy), f0z = floorf(fz);
        const float wx = fx - f0x, wy = fy - f0y, wz = fz - f0z;
        const float mwx = 1.f - wx, mwy = 1.f - wy, mwz = 1.f - wz;
        const unsigned ix = (unsigned)f0x;
        const unsigned iy = (unsigned)f0y;
        const unsigned iz = (unsigned)f0z;
        // PRIMES[0] == 1, so hx contributes raw ix; incremental +PRIME for the +1 corners
        const unsigned hy0 = iy * P1, hy1 = hy0 + P1;
        const unsigned hz0 = iz * P2, hz1 = hz0 + P2;
        const float* tab = table + (size_t)lv * ((size_t)1u << 19) * 2u;

        const float wyz00 = mwy * mwz, wyz01 = mwy * wz;
        const float wyz10 = wy  * mwz, wyz11 = wy  * wz;

        float f0 = 0.f, f1 = 0.f;
        #pragma unroll
        for (int ci = 0; ci < 8; ++ci) {
            const int i = (ci >> 2) & 1, j = (ci >> 1) & 1, k = ci & 1;
            const unsigned h = (ix + (unsigned)i) ^ (j ? hy1 : hy0) ^ (k ? hz1 : hz0);
            const unsigned idx = h & TMASK;
            const v2f t = *(const v2f*)(tab + (size_t)idx * 2u);   // global_load_b64 gather (L2-resident)
            const float cw = (i ? wx : mwx) *
                             (j ? (k ? wyz11 : wyz10) : (k ? wyz01 : wyz00));
            f0 = fmaf(cw, t.x, f0);
            f1 = fmaf(cw, t.y, f1);
        }
        *(v2f*)(myFeat + 2 * lv) = (v2f){f0, f1};   // ds_store_b64, aligned (PAD*4 % 8 == 0)
    }

    __syncthreads();

    // ---- Phase 2: 32->16(3 used) linear via V_WMMA_F32_16X16X4_F32, bias folded into C ----
    const int lane = tid & 31;
    const int wave = tid >> 5;
    const int m    = lane & 15;               // matrix row within tile / output column id
    const int koff = (lane >= 16) ? 2 : 0;    // A-fragment K split per lane half
    const int roff = (lane >= 16) ? 8 : 0;    // D-fragment row split per lane half
    const float bn = (m < 3) ? bias[m] : 0.f;

    #pragma unroll
    for (int t = 0; t < 2; ++t) {
        const int rowBase = wave * 32 + t * 16;   // local point row base within block
        v8f c;
        #pragma unroll
        for (int v = 0; v < 8; ++v) c[v] = bn;    // bias pre-loaded into accumulator

        #pragma unroll
        for (int j = 0; j < 8; ++j) {
            // A 16x4 f32: lanes0-15 hold K=4j+0/1, lanes16-31 hold K=4j+2/3 (VGPR0/1)
            const v2f a = *(const v2f*)(&sFeat[(rowBase + m) * PAD + 4 * j + koff]);
            const v2f b = sB[j * 32 + lane];
            c = __builtin_amdgcn_wmma_f32_16x16x4_f32(
                    /*neg_a=*/false, a, /*neg_b=*/false, b,
                    /*c_mod=*/(short)0, c, /*reuse_a=*/false, /*reuse_b=*/false);
        }

        // D layout: VGPR v -> row rowBase+v+roff, column m (= lane & 15).
        // Stage the 3 live columns in LDS (within-wave dependence only).
        if (m < 3) {
            #pragma unroll
            for (int v = 0; v < 8; ++v) {
                sOut[(rowBase + v + roff) * 3 + m] = c[v];
            }
        }
    }

    // ---- Coalesced writeback: one 12-byte row per thread ----
    {
        const int row = blockIdx.x * 256 + tid;
        if (row < N) {
            float3 o;
            o.x = sOut[tid * 3 + 0];
            o.y = sOut[tid * 3 + 1];
            o.z = sOut[tid * 3 + 2];
            out[row * 3 + 0] = o.x;
            out[row * 3 + 1] = o.y;
            out[row * 3 + 2] = o.z;
        }
    }
}

extern "C" void kernel_launch(void* const* d_in, const int* in_sizes, int n_in,
                              void* d_out, int out_size, void* d_ws, size_t ws_size,
                              hipStream_t stream) {
    const float* x     = (const float*)d_in[0];   // [N,3]
    const float* table = (const float*)d_in[1];   // [16, 2^19, 2]
    const float* W     = (const float*)d_in[2];   // [3, 32]
    const float* b     = (const float*)d_in[3];   // [3]
    float* out = (float*)d_out;                   // [N,3]
    const int N = in_sizes[0] / 3;
    const int blocks = (N + 255) / 256;
    hipLaunchKernelGGL(hashgrid_wmma_kernel, dim3(blocks), dim3(256), 0, stream,
                       x, table, W, b, out, N);
}